// LTM_6184752906438
// MI455X (gfx1250) — compile-verified
//
#include <hip/hip_runtime.h>

// ---------------------------------------------------------------------------
// Local tone mapping (4K) for MI455X / gfx1250.
// Bandwidth-bound pipeline; 17-tap separable box filters of the guided filter
// are expressed as banded-matrix GEMMs on V_WMMA_F32_16X16X4_F32 (wave32).
// Vertical pass stages tiles through LDS (transposed, bank-conflict-free).
// ---------------------------------------------------------------------------

typedef float v2f __attribute__((ext_vector_type(2)));
typedef float v8f __attribute__((ext_vector_type(8)));

namespace {
constexpr int IMG_H = 2160;
constexpr int IMG_W = 3840;
constexpr int DNH   = 1080;   // IMG_H * 0.5
constexpr int DNW   = 1920;   // IMG_W * 0.5
constexpr int RAD   = 8;      // box radius -> 17 taps
constexpr float WBOX      = 1.0f / 17.0f;
constexpr float EPS_GF    = 1e-3f;
constexpr float EPS_LOG   = 1e-6f;
constexpr float EPS_SCALE = 1e-4f;
constexpr float A_COEF    = 0.7f;
constexpr int LDS_PITCH = 33;  // 32 rows + 1 pad: 33 invertible mod 64 banks
}

static __device__ __forceinline__ int refl(int i, int n) {
  i = (i < 0) ? -i : i;                 // jnp.pad 'reflect' (no edge repeat)
  return (i >= n) ? (2 * n - 2 - i) : i;
}
static __device__ __forceinline__ int imin(int a, int b) { return a < b ? a : b; }

// D = A(16x4,f32) * B(4x16,f32) + C(16x16,f32), wave32, full EXEC.
static __device__ __forceinline__ v8f wmma4(v2f a, v2f b, v8f c) {
  return __builtin_amdgcn_wmma_f32_16x16x4_f32(
      /*neg_a=*/false, a, /*neg_b=*/false, b,
      /*c_mod=*/(short)0, c, /*reuse_a=*/false, /*reuse_b=*/false);
}

// Two adjacent taps of the banded box matrix for offsets (t, t+1).
static __device__ __forceinline__ v2f band2(unsigned band, int t) {
  return (v2f){((band >> t) & 1u) ? WBOX : 0.0f,
               ((band >> (t + 1)) & 1u) ? WBOX : 0.0f};
}

static __device__ __forceinline__ float luma_log2_at(const float* __restrict__ x, size_t p) {
  float r = x[3 * p], g = x[3 * p + 1], b = x[3 * p + 2];
  float Y = 0.2126f * r + 0.7152f * g + 0.0722f * b;
  return log2f(fmaxf(Y, EPS_LOG));
}

// ---------------------------------------------------------------------------
// Kernel 1: fused luminance -> log2 -> exact 2x bilinear downsample.
// (scale==2, align_corners=False => plain 2x2 average of log-luma)
// ---------------------------------------------------------------------------
__global__ __launch_bounds__(256) void down_luma(const float* __restrict__ x,
                                                 float* __restrict__ yld) {
  int p = blockIdx.x * 256 + threadIdx.x;
  if (p >= DNH * DNW) return;
  int i = p / DNW, j = p % DNW;
  size_t r0 = (size_t)(2 * i) * IMG_W + (size_t)(2 * j);
  float a = luma_log2_at(x, r0);
  float b = luma_log2_at(x, r0 + 1);
  float c = luma_log2_at(x, r0 + IMG_W);
  float d = luma_log2_at(x, r0 + IMG_W + 1);
  yld[p] = 0.25f * (a + b + c + d);
}

// ---------------------------------------------------------------------------
// Kernel 2/4: horizontal 17-tap box via WMMA.
//   Out[16x16] = In[16x32] x Band[32x16],  Band[t][n] = 1/17 iff n <= t <= n+16
// One wave (32 threads) per 16x16 output tile; 8 K-steps of K=4.
// Interior tiles use aligned float2 loads; only the 2 edge tile-columns take
// the reflect-indexed scalar path (uniform branch).
// SQUARE=true: second chain filters In^2 (for mean_II).
// ---------------------------------------------------------------------------
template <bool SQUARE>
__global__ __launch_bounds__(32) void hbox(const float* __restrict__ s0,
                                           const float* __restrict__ s1,
                                           float* __restrict__ d0,
                                           float* __restrict__ d1) {
  const int lane = threadIdx.x;
  const int m  = lane & 15;       // A row / D column index
  const int hh = lane >> 4;       // wave half
  const int TX = DNW / 16;
  const int c0 = (blockIdx.x % TX) * 16;
  const int r0 = (blockIdx.x / TX) * 16;
  const int row = refl(r0 + m, DNH);
  const size_t rowoff = (size_t)row * DNW;
  const unsigned band = 0x1FFFFu << m;   // taps m..m+16 of the 32-wide window

  v8f acc0 = {0.f, 0.f, 0.f, 0.f, 0.f, 0.f, 0.f, 0.f};
  v8f acc1 = {0.f, 0.f, 0.f, 0.f, 0.f, 0.f, 0.f, 0.f};

  if (c0 >= RAD && c0 + 16 + RAD <= DNW) {     // interior (uniform branch)
    const float* p0 = s0 + rowoff + (c0 - RAD);
    const float* p1 = s1 + rowoff + (c0 - RAD);
#pragma unroll
    for (int s = 0; s < 8; ++s) {
      const int t = 4 * s + 2 * hh;            // even => 8B-aligned pair
      const v2f A = *reinterpret_cast<const v2f*>(p0 + t);
      const v2f A2 = SQUARE ? (v2f){A.x * A.x, A.y * A.y}
                            : *reinterpret_cast<const v2f*>(p1 + t);
      const v2f B = band2(band, t);
      acc0 = wmma4(A, B, acc0);
      acc1 = wmma4(A2, B, acc1);
    }
  } else {                                     // edge tiles: reflect columns
#pragma unroll
    for (int s = 0; s < 8; ++s) {
      const int t = 4 * s + 2 * hh;
      const int ca = refl(c0 - RAD + t, DNW);
      const int cb = refl(c0 - RAD + t + 1, DNW);
      const float a0 = s0[rowoff + ca];
      const float a1 = s0[rowoff + cb];
      const v2f A = {a0, a1};
      const v2f A2 = SQUARE ? (v2f){a0 * a0, a1 * a1}
                            : (v2f){s1[rowoff + ca], s1[rowoff + cb]};
      const v2f B = band2(band, t);
      acc0 = wmma4(A, B, acc0);
      acc1 = wmma4(A2, B, acc1);
    }
  }

  const size_t obase = (size_t)(r0 + 8 * hh) * DNW + c0 + m;
  if (r0 + 16 <= DNH) {                        // full tile (uniform branch)
#pragma unroll
    for (int v = 0; v < 8; ++v) {
      d0[obase + (size_t)v * DNW] = acc0[v];
      d1[obase + (size_t)v * DNW] = acc1[v];
    }
  } else {
#pragma unroll
    for (int v = 0; v < 8; ++v) {
      if (r0 + v + 8 * hh < DNH) {
        d0[obase + (size_t)v * DNW] = acc0[v];
        d1[obase + (size_t)v * DNW] = acc1[v];
      }
    }
  }
}

// ---------------------------------------------------------------------------
// Kernel 3/5: vertical 17-tap box via WMMA + fused guided-filter elementwise.
//   Out[16x16] = Band[16x32] x In[32x16]
// 32x16 input tiles of both operands are staged into LDS transposed
// (column-major, pitch 33 -> bank-conflict-free stores and loads), turning
// the strided global row reads into coalesced 64B bursts + ds reads.
// MODE 0: inputs (hI, hII) -> outputs (a, b)
// MODE 1: inputs (ha, hb) + yld -> output base = mean_a*I + mean_b
// ---------------------------------------------------------------------------
template <int MODE>
__global__ __launch_bounds__(32) void vbox(const float* __restrict__ s0,
                                           const float* __restrict__ s1,
                                           const float* __restrict__ yld,
                                           float* __restrict__ d0,
                                           float* __restrict__ d1) {
  __shared__ float ldsA[16 * LDS_PITCH];
  __shared__ float ldsB[16 * LDS_PITCH];

  const int lane = threadIdx.x;
  const int m  = lane & 15;
  const int hh = lane >> 4;
  const int TX = DNW / 16;
  const int c0 = (blockIdx.x % TX) * 16;
  const int r0 = (blockIdx.x / TX) * 16;
  const int col = c0 + m;                 // B/D column this lane owns

  // Stage: 32 rows x 16 cols, coalesced (half-wave = 64B row burst),
  // stored transposed: lds[col * 33 + row].
  {
    const int cc = lane & 15;
    const int rb = lane >> 4;
#pragma unroll
    for (int it = 0; it < 16; ++it) {
      const int rloc = 2 * it + rb;                       // 0..31
      const int rg = refl(r0 - RAD + rloc, DNH);
      const size_t go = (size_t)rg * DNW + c0 + cc;
      ldsA[cc * LDS_PITCH + rloc] = s0[go];
      ldsB[cc * LDS_PITCH + rloc] = s1[go];
    }
  }
  // Single wave per workgroup: LDS ops are in-order; no barrier required.

  v8f acc0 = {0.f, 0.f, 0.f, 0.f, 0.f, 0.f, 0.f, 0.f};
  v8f acc1 = {0.f, 0.f, 0.f, 0.f, 0.f, 0.f, 0.f, 0.f};
  const unsigned band = 0x1FFFFu << m;    // A[m][t] = 1/17 iff m <= t <= m+16
  const float* colA = &ldsA[m * LDS_PITCH];
  const float* colB = &ldsB[m * LDS_PITCH];

#pragma unroll
  for (int s = 0; s < 8; ++s) {
    const int t = 4 * s + 2 * hh;
    const v2f A  = band2(band, t);
    const v2f B0 = {colA[t], colA[t + 1]};
    const v2f B1 = {colB[t], colB[t + 1]};
    acc0 = wmma4(A, B0, acc0);
    acc1 = wmma4(A, B1, acc1);
  }

  const size_t obase = (size_t)(r0 + 8 * hh) * DNW + col;
#pragma unroll
  for (int v = 0; v < 8; ++v) {
    const int rr = r0 + v + 8 * hh;
    if (rr < DNH) {
      const size_t o = obase + (size_t)v * DNW;
      const float m0 = acc0[v];           // mean_I   | mean_a
      const float m1 = acc1[v];           // mean_II  | mean_b
      if (MODE == 0) {
        const float var = m1 - m0 * m0;
        const float a = var / (var + EPS_GF);
        d0[o] = a;
        d1[o] = m0 - a * m0;
      } else {
        d0[o] = m0 * yld[o] + m1;         // base_down
      }
    }
  }
}

// ---------------------------------------------------------------------------
// Kernel 6: bilinear 2x upsample of base + recombine + exp2 + scale + clamp.
// ---------------------------------------------------------------------------
__global__ __launch_bounds__(256) void finalize_k(const float* __restrict__ x,
                                                  const float* __restrict__ based,
                                                  float* __restrict__ out) {
  int p = blockIdx.x * 256 + threadIdx.x;
  if (p >= IMG_H * IMG_W) return;
  __builtin_prefetch(x + 3 * (size_t)p + 6144, 0, 0);  // global_prefetch_b8

  const int i = p / IMG_W, j = p % IMG_W;
  // torch align_corners=False: src = (dst+0.5)*0.5 - 0.5, clipped.
  const float ys = fminf(fmaxf(0.5f * (float)i - 0.25f, 0.0f), (float)(DNH - 1));
  const float xs = fminf(fmaxf(0.5f * (float)j - 0.25f, 0.0f), (float)(DNW - 1));
  const int y0 = (int)ys, x0i = (int)xs;
  const float wy = ys - (float)y0, wx = xs - (float)x0i;
  const int y1 = imin(y0 + 1, DNH - 1), x1 = imin(x0i + 1, DNW - 1);
  const float Ia = based[(size_t)y0 * DNW + x0i];
  const float Ib = based[(size_t)y0 * DNW + x1];
  const float Ic = based[(size_t)y1 * DNW + x0i];
  const float Id = based[(size_t)y1 * DNW + x1];
  const float base = (Ia + (Ib - Ia) * wx) * (1.0f - wy) + (Ic + (Id - Ic) * wx) * wy;

  const float r = x[3 * (size_t)p], g = x[3 * (size_t)p + 1], b = x[3 * (size_t)p + 2];
  const float Y = 0.2126f * r + 0.7152f * g + 0.0722f * b;
  const float yl = log2f(fmaxf(Y, EPS_LOG));
  const float tm = A_COEF * base + (yl - base);      // base*A + detail
  const float sc = exp2f(tm) / (Y + EPS_SCALE);
  out[3 * (size_t)p]     = fminf(fmaxf(r * sc, 0.0f), 1.0f);
  out[3 * (size_t)p + 1] = fminf(fmaxf(g * sc, 0.0f), 1.0f);
  out[3 * (size_t)p + 2] = fminf(fmaxf(b * sc, 0.0f), 1.0f);
}

// ---------------------------------------------------------------------------
extern "C" void kernel_launch(void* const* d_in, const int* in_sizes, int n_in,
                              void* d_out, int out_size, void* d_ws, size_t ws_size,
                              hipStream_t stream) {
  (void)in_sizes; (void)n_in; (void)out_size; (void)ws_size;
  const float* x = (const float*)d_in[0];
  float* out = (float*)d_out;

  // Workspace: 5 quarter-res planes (~41.5 MB total) — fully L2-resident.
  const size_t DNPIX = (size_t)DNH * DNW;
  float* ws  = (float*)d_ws;
  float* yld = ws;              // log-luma, downsampled
  float* t0  = yld + DNPIX;     // hI  / ha
  float* t1  = t0 + DNPIX;      // hII / hb
  float* t2  = t1 + DNPIX;      // a   / base_down
  float* t3  = t2 + DNPIX;      // b

  const int npix   = IMG_H * IMG_W;
  const int ndown  = DNH * DNW;
  const int ntiles = (DNW / 16) * ((DNH + 15) / 16);   // 120 * 68 tiles

  down_luma<<<(ndown + 255) / 256, 256, 0, stream>>>(x, yld);
  hbox<true ><<<ntiles, 32, 0, stream>>>(yld, yld, t0, t1);          // hI, hII
  vbox<0>    <<<ntiles, 32, 0, stream>>>(t0, t1, yld, t2, t3);       // a, b
  hbox<false><<<ntiles, 32, 0, stream>>>(t2, t3, t0, t1);            // ha, hb
  vbox<1>    <<<ntiles, 32, 0, stream>>>(t0, t1, yld, t2, t3);       // base_down -> t2
  finalize_k<<<(npix + 255) / 256, 256, 0, stream>>>(x, t2, out);
}